// compNormSimiMatrix_batch_14499809591880
// MI455X (gfx1250) — compile-verified
//
#include <hip/hip_runtime.h>
#include <stdint.h>

typedef __attribute__((ext_vector_type(2))) float v2f;
typedef __attribute__((ext_vector_type(4))) float v4f;
typedef __attribute__((ext_vector_type(8))) float v8f;

#define ROWS_PER_BLOCK 16
#define COLS           2048
#define LDS_STRIDE     2052   // 2048 + 4 pad floats -> stride%64==4 -> conflict-free transposed reads
#define THREADS        256
#define CHUNKS         32     // 16*2048 floats / (4 floats * 256 threads)

__global__ __launch_bounds__(THREADS)
void rownorm_wmma_async_kernel(const float* __restrict__ in, float* __restrict__ out) {
    extern __shared__ float smem[];
    float* tile    = smem;                                  // 16 x 2052 floats
    float* partial = smem + ROWS_PER_BLOCK * LDS_STRIDE;    // 8 waves x 16 rows
    float* invv    = partial + 8 * 16;                      // 16 floats

    const int tid  = threadIdx.x;
    const int lane = tid & 31;
    const int w    = tid >> 5;                              // wave id 0..7
    const size_t rowBase = (size_t)blockIdx.x * ROWS_PER_BLOCK;

    // ---- Phase 1: async DMA global -> LDS (bypasses VGPRs, ASYNCcnt-tracked) ----
    #pragma unroll
    for (int i = 0; i < CHUNKS; ++i) {
        const int c   = tid + THREADS * i;   // 16B-chunk index in tile
        const int row = c >> 9;              // 512 chunks per row
        const int col = (c & 511) << 2;      // float index within row
        const uint32_t ldsb = (uint32_t)(uintptr_t)&tile[row * LDS_STRIDE + col];
        const float* g = in + (rowBase + row) * (size_t)COLS + col;
        asm volatile("global_load_async_to_lds_b128 %0, %1, off"
                     :: "v"(ldsb), "v"(g) : "memory");
    }
    asm volatile("s_wait_asynccnt 0" ::: "memory");
    __syncthreads();

    // ---- Phase 2: WMMA row-sum. D = A x ones + C accumulates 16 row sums. ----
    {
        const int rowA = lane & 15;                 // A layout: M = lane % 16
        const int koff = (lane >> 4) << 1;          // lanes 16..31 hold K+2,K+3
        const float* rowp = tile + rowA * LDS_STRIDE + koff;

        v2f bones; bones.x = 1.0f; bones.y = 1.0f;  // B = ones(4x16), layout-invariant
        v8f acc = {};

        const int k0 = w * (COLS / 8);              // 256 columns per wave
        #pragma unroll 4
        for (int k = 0; k < COLS / 8; k += 4) {
            v2f a = *(const v2f*)(rowp + k0 + k);   // ds_load_b64, conflict-free
            acc = __builtin_amdgcn_wmma_f32_16x16x4_f32(
                      /*neg_a=*/false, a, /*neg_b=*/false, bones,
                      /*c_mod=*/(short)0, acc, /*reuse_a=*/false, /*reuse_b=*/false);
        }

        // C/D layout: vgpr v, lanes 0-15 -> M=v ; lanes 16-31 -> M=v+8 (columns identical)
        if (lane == 0) {
            #pragma unroll
            for (int v = 0; v < 8; ++v) partial[w * 16 + v] = acc[v];
        } else if (lane == 16) {
            #pragma unroll
            for (int v = 0; v < 8; ++v) partial[w * 16 + 8 + v] = acc[v];
        }
    }
    __syncthreads();

    if (tid < 16) {
        float s = 0.0f;
        #pragma unroll
        for (int ww = 0; ww < 8; ++ww) s += partial[ww * 16 + tid];
        invv[tid] = 1.0f / fmaxf(s, 1e-5f);
    }
    __syncthreads();

    // ---- Phase 3: scale from LDS, non-temporal B128 stores to global ----
    #pragma unroll
    for (int i = 0; i < CHUNKS; ++i) {
        const int c   = tid + THREADS * i;
        const int row = c >> 9;
        const int col = (c & 511) << 2;
        v4f v = *(const v4f*)&tile[row * LDS_STRIDE + col];
        v *= invv[row];
        __builtin_nontemporal_store(v, (v4f*)(out + (rowBase + row) * (size_t)COLS + col));
    }
}

extern "C" void kernel_launch(void* const* d_in, const int* in_sizes, int n_in,
                              void* d_out, int out_size, void* d_ws, size_t ws_size,
                              hipStream_t stream) {
    const float* in = (const float*)d_in[0];
    float* out = (float*)d_out;

    const int totalRows = in_sizes[0] / COLS;          // 16 * 2048 = 32768
    const int nBlocks   = totalRows / ROWS_PER_BLOCK;  // 2048

    const size_t shmem = (ROWS_PER_BLOCK * LDS_STRIDE + 8 * 16 + 16) * sizeof(float);
    rownorm_wmma_async_kernel<<<nBlocks, THREADS, shmem, stream>>>(in, out);
}